// ChebConvLayer_10943576670987
// MI455X (gfx1250) — compile-verified
//
#include <hip/hip_runtime.h>
#include <hip/hip_bf16.h>

// ---------------------------------------------------------------------------
// ChebConv (K=4, sym norm, lambda_max=2) + bias + BatchNorm1d + LeakyReLU
// MI455X / gfx1250: fp32 WMMA (v_wmma_f32_16x16x4_f32) for the dense T_k @ W_k
// GEMMs; L2-resident atomic scatter-add for the sparse L_hat propagation.
// ---------------------------------------------------------------------------

typedef float v2f __attribute__((ext_vector_type(2)));
typedef float v8f __attribute__((ext_vector_type(8)));

#define FEAT 128
#define BN_EPS_F 1e-5f
#define LEAKY_F 0.01f

// ---------------- elementwise helpers ----------------

__global__ void fill_f32_k(float* __restrict__ p, float v, long long n) {
    long long i = (long long)blockIdx.x * blockDim.x + threadIdx.x;
    if (i < n) p[i] = v;
}

// dst[i] = s * src[i]  (safe in-place: pure elementwise)
__global__ void scaled_copy_k(float* __restrict__ dst, const float* __restrict__ src,
                              float s, long long n) {
    long long i = (long long)blockIdx.x * blockDim.x + threadIdx.x;
    if (i < n) dst[i] = s * src[i];
}

// ---------------- graph preprocessing ----------------

__global__ void degree_k(const int* __restrict__ row, float* __restrict__ deg, int E) {
    int e = blockIdx.x * blockDim.x + threadIdx.x;
    if (e < E) atomicAdd(&deg[row[e]], 1.0f);
}

__global__ void dinv_k(float* __restrict__ deg, int N) {
    int i = blockIdx.x * blockDim.x + threadIdx.x;
    if (i < N) {
        float d = deg[i];
        deg[i] = (d > 0.0f) ? rsqrtf(d) : 0.0f;
    }
}

__global__ void edgew_k(const int* __restrict__ row, const int* __restrict__ col,
                        const float* __restrict__ dinv, float* __restrict__ ew, int E) {
    int e = blockIdx.x * blockDim.x + threadIdx.x;
    if (e < E) ew[e] = -dinv[row[e]] * dinv[col[e]];
}

// ---------------- sparse propagation: dst[col] += scale * ew * src[row] ------
// One wave32 per edge; each lane owns 4 contiguous features (128 = 32*4).
// Gather is a coalesced b128 load; scatter is 4 global_atomic_add_f32.
// Working set (~190 MB) is ~L2 resident on MI455X (192 MB), so the random
// gathers/atomics mostly resolve in L2.

__global__ void scatter_k(const float* __restrict__ src, float* __restrict__ dst,
                          const float* __restrict__ ew,
                          const int* __restrict__ row, const int* __restrict__ col,
                          int E, float scale) {
    int e = blockIdx.x * 8 + (threadIdx.x >> 5);  // 8 edges per 256-thread block
    if (e >= E) return;
    int lane = threadIdx.x & 31;
    float w = scale * ew[e];
    const float4* s = (const float4*)(src + (size_t)row[e] * FEAT) + lane;
    float4 v = *s;
    float* d = dst + (size_t)col[e] * FEAT + lane * 4;
    atomicAdd(d + 0, w * v.x);
    atomicAdd(d + 1, w * v.y);
    atomicAdd(d + 2, w * v.z);
    atomicAdd(d + 3, w * v.w);
}

// ---------------- dense GEMM via fp32 WMMA ----------------
// out[M,128] (+)= A[M,128] @ B[128,128] (+ bias).
// Block = 256 threads = 8 waves; wave w owns the 16x16 tile at cols [16w,16w+16).
// Per ISA 7.12.2 (wave32):
//   A 16x4 f32 : lanes 0-15 rows M=0..15 with {K=k0,k0+1}; lanes 16-31 {k0+2,k0+3}
//   B 4x16 f32 : lane li cols N=li with rows {k0,k0+1} / {k0+2,k0+3} by half
//   C/D 16x16  : VGPR r -> M = r + 8*(lane/16), N = lane%16

__global__ void gemm_wmma_k(const float* __restrict__ A, const float* __restrict__ B,
                            const float* __restrict__ bias, float* __restrict__ out,
                            int nrows, int accumulate) {
    const int wave = threadIdx.x >> 5;
    const int lane = threadIdx.x & 31;
    const int half = lane >> 4;       // 0: lanes 0-15, 1: lanes 16-31
    const int li   = lane & 15;
    const int row0 = blockIdx.x * 16;
    const int col0 = wave * 16;

    // clamp A row for (potential) ragged tail; keeps EXEC fully on for WMMA
    int arow = row0 + li;
    if (arow > nrows - 1) arow = nrows - 1;
    const float* ap = A + (size_t)arow * FEAT;
    const float* bp = B + col0 + li;

    v8f acc = {};
#pragma unroll
    for (int k0 = 0; k0 < FEAT; k0 += 4) {
        const int ka = k0 + 2 * half;
        v2f a = *(const v2f*)(ap + ka);           // A[arow][ka], A[arow][ka+1]
        v2f b;
        b.x = bp[(size_t)ka * FEAT];              // B[ka][col]
        b.y = bp[(size_t)(ka + 1) * FEAT];        // B[ka+1][col]
        // (neg_a, A, neg_b, B, c_mod, C, reuse_a, reuse_b)
        acc = __builtin_amdgcn_wmma_f32_16x16x4_f32(
            false, a, false, b, (short)0, acc, false, false);
    }

    const int col   = col0 + li;
    const int mbase = row0 + 8 * half;
    const float bv  = bias ? bias[col] : 0.0f;
#pragma unroll
    for (int r = 0; r < 8; ++r) {
        int m = mbase + r;
        if (m < nrows) {
            size_t idx = (size_t)m * FEAT + col;
            float v = acc[r] + bv;
            if (accumulate) v += out[idx];
            out[idx] = v;
        }
    }
}

// ---------------- BatchNorm statistics + apply ----------------

__global__ void bn_stats_k(const float* __restrict__ out, float* __restrict__ stats,
                           int N) {
    int t = blockIdx.x * blockDim.x + threadIdx.x;
    int c = t & (FEAT - 1);
    int r0 = t >> 7;
    int rs = (gridDim.x * blockDim.x) >> 7;
    float s = 0.0f, s2 = 0.0f;
    for (int r = r0; r < N; r += rs) {
        float v = out[(size_t)r * FEAT + c];
        s += v;
        s2 += v * v;
    }
    atomicAdd(&stats[c], s);
    atomicAdd(&stats[FEAT + c], s2);
}

__global__ void bn_apply_k(float* __restrict__ out, const float* __restrict__ stats,
                           const float* __restrict__ gamma, const float* __restrict__ beta,
                           int N) {
    long long i = (long long)blockIdx.x * blockDim.x + threadIdx.x;
    long long total = (long long)N * FEAT;
    if (i >= total) return;
    int c = (int)(i & (FEAT - 1));
    float inv_n = 1.0f / (float)N;
    float mean = stats[c] * inv_n;
    float var  = stats[FEAT + c] * inv_n - mean * mean;
    float xn = (out[i] - mean) * rsqrtf(var + BN_EPS_F);
    float y = xn * gamma[c] + beta[c];
    out[i] = (y >= 0.0f) ? y : LEAKY_F * y;
}

// ---------------- driver ----------------

extern "C" void kernel_launch(void* const* d_in, const int* in_sizes, int n_in,
                              void* d_out, int out_size, void* d_ws, size_t ws_size,
                              hipStream_t stream) {
    const float* x     = (const float*)d_in[0];
    const int*   eidx  = (const int*)d_in[1];   // int32 (JAX default x64-off)
    const float* W     = (const float*)d_in[2]; // [K,128,128]
    const float* bias  = (const float*)d_in[3];
    const float* gamma = (const float*)d_in[4];
    const float* beta  = (const float*)d_in[5];
    float* out = (float*)d_out;

    const int N = in_sizes[0] / FEAT;
    const int E = in_sizes[1] / 2;
    const int K = in_sizes[2] / (FEAT * FEAT);
    const int* row = eidx;
    const int* col = eidx + E;
    const long long NF = (long long)N * FEAT;

    // workspace carve-up (floats): deg/dinv | ew | bufA | bufB | stats
    float* ws = (float*)d_ws;
    size_t off = 0;
    float* deg   = ws + off; off += ((size_t)N + 15) & ~(size_t)15;
    float* ew    = ws + off; off += ((size_t)E + 15) & ~(size_t)15;
    float* bufA  = ws + off; off += (size_t)NF;
    float* bufB  = ws + off; off += (size_t)NF;
    float* stats = ws + off; off += 2 * FEAT;
    (void)ws_size;

    auto cdiv = [](long long a, long long b) { return (int)((a + b - 1) / b); };

    // 1) deg -> dinv -> edge weights of L_hat
    fill_f32_k<<<cdiv(N, 256), 256, 0, stream>>>(deg, 0.0f, N);
    degree_k<<<cdiv(E, 256), 256, 0, stream>>>(row, deg, E);
    dinv_k<<<cdiv(N, 256), 256, 0, stream>>>(deg, N);
    edgew_k<<<cdiv(E, 256), 256, 0, stream>>>(row, col, deg, ew, E);

    const int mtiles = cdiv(N, 16);

    // 2) out = T0 @ W0 + b   (T0 = x)
    gemm_wmma_k<<<mtiles, 256, 0, stream>>>(x, W, bias, out, N, 0);

    if (K > 1) {
        // 3) T1 = L_hat x  (zero-init + scatter), out += T1 @ W1
        fill_f32_k<<<cdiv(NF, 256), 256, 0, stream>>>(bufA, 0.0f, NF);
        scatter_k<<<cdiv(E, 8), 256, 0, stream>>>(x, bufA, ew, row, col, E, 1.0f);
        gemm_wmma_k<<<mtiles, 256, 0, stream>>>(bufA, W + (size_t)FEAT * FEAT,
                                                nullptr, out, N, 1);

        // 4) T_k = 2 L_hat T_{k-1} - T_{k-2}: init dst = -T_{k-2}, scatter 2*ew*T_{k-1}
        const float* Tkm2 = x;
        float* Tkm1 = bufA;
        float* cur  = bufB;
        for (int k = 2; k < K; ++k) {
            scaled_copy_k<<<cdiv(NF, 256), 256, 0, stream>>>(cur, Tkm2, -1.0f, NF);
            scatter_k<<<cdiv(E, 8), 256, 0, stream>>>(Tkm1, cur, ew, row, col, E, 2.0f);
            gemm_wmma_k<<<mtiles, 256, 0, stream>>>(cur, W + (size_t)k * FEAT * FEAT,
                                                    nullptr, out, N, 1);
            float* spare = Tkm1;   // slot that becomes writable next iteration
            Tkm2 = Tkm1;
            Tkm1 = cur;
            cur  = spare;          // cur == Tkm2 afterwards -> in-place negate is fine
        }
    }

    // 5) BatchNorm (batch stats) + LeakyReLU, in place on d_out
    fill_f32_k<<<1, 256, 0, stream>>>(stats, 0.0f, 2 * FEAT);
    bn_stats_k<<<512, 256, 0, stream>>>(out, stats, N);
    bn_apply_k<<<cdiv(NF, 256), 256, 0, stream>>>(out, stats, gamma, beta, N);
}